// DSRQS_31894427140771
// MI455X (gfx1250) — compile-verified
//
#include <hip/hip_runtime.h>
#include <math.h>

typedef float v2f __attribute__((ext_vector_type(2)));
typedef float v8f __attribute__((ext_vector_type(8)));
typedef unsigned int u32x4 __attribute__((ext_vector_type(4)));
typedef int i32x4 __attribute__((ext_vector_type(4)));
typedef int i32x8 __attribute__((ext_vector_type(8)));

#define D 128
#define RHO 8
#define NHOP 5
#define WAVES 8
#define ROWS_PER_WAVE 16
#define ROWS_PER_BLOCK (WAVES * ROWS_PER_WAVE)   // 128
#define W_STRIDE 144   // row-major W0 rows padded by TDM: 144 % 32 == 16 -> the two
                       // half-wave groups (2 rows = 288 dw = bank+32) hit disjoint banks
#define T_STRIDE 129   // T tile fits inside the (dead) W0 region

// Shared memory layout (floats). sT aliases sW (W0 dead after GEMM phase).
#define OFF_W    0
#define SZ_W     (D * W_STRIDE)                   // 18432
#define OFF_A    (OFF_W + SZ_W)
#define SZ_A     (NHOP * D * RHO)                 // 5120
#define OFF_B    (OFF_A + SZ_A)
#define SZ_B     SZ_A
#define OFF_V    (OFF_B + SZ_B)
#define SZ_V     D
#define OFF_BIAS (OFF_V + SZ_V)
#define SZ_BIAS  8
#define SMEM_FLOATS (OFF_BIAS + SZ_BIAS)          // 28808 floats = 115,232 B
// per-wave T tiles (aliased at OFF_W): WAVES*16*T_STRIDE = 16512 <= SZ_W (18432)

// ---- Tensor DMA descriptor packing (CDNA5 D#, CDNA5 ISA) ----
__device__ __forceinline__ u32x4 tdm_group0(unsigned lds_addr, const void* gaddr) {
    unsigned long long ga = (unsigned long long)gaddr;
    u32x4 g0;
    g0.x = 1u;                                   // count=1, is_restore=0, gather=0
    g0.y = lds_addr;                             // bits[63:32] lds byte address
    g0.z = (unsigned)(ga & 0xFFFFFFFFu);         // global_addr[31:0]
    g0.w = (unsigned)((ga >> 32) & 0x01FFFFFFu)  // global_addr[56:32]
         | (2u << 30);                           // type=2 ("image")
    return g0;
}
__device__ __forceinline__ i32x8 tdm_group1(unsigned flags,
                                            unsigned td0, unsigned td1,
                                            unsigned tile0, unsigned tile1,
                                            unsigned long long stride0) {
    i32x8 g1;
    g1[0] = (int)flags;                                         // mask=0,data_size,pad ctl
    g1[1] = (int)((td0 & 0xFFFFu) << 16);                       // abar=0 | td0 lo16
    g1[2] = (int)((td0 >> 16) | ((td1 & 0xFFFFu) << 16));       // td0 hi16 | td1 lo16
    g1[3] = (int)((td1 >> 16) | (tile0 << 16));                 // td1 hi16 | tile_dim0
    g1[4] = (int)(tile1 & 0xFFFFu);                             // tile_dim1 | tile_dim2=0
    g1[5] = (int)(unsigned)(stride0 & 0xFFFFFFFFu);             // dim0_stride lo32
    g1[6] = (int)(unsigned)(stride0 >> 32);                     // dim0_stride hi16 | s1 lo
    g1[7] = 0;                                                  // dim1_stride hi
    return g1;
}

__global__ __launch_bounds__(256) void DSRQS_fused_kernel(
    const float* __restrict__ q, const float* __restrict__ r,
    const int* __restrict__ hop, const float* __restrict__ W0,
    const float* __restrict__ A, const float* __restrict__ Bm,
    const float* __restrict__ v, const float* __restrict__ bvec,
    float* __restrict__ out)
{
    extern __shared__ float smem[];
    float* sW    = smem + OFF_W;     // W0 row-major, row stride W_STRIDE (TDM-padded)
    float* sA    = smem + OFF_A;
    float* sB    = smem + OFF_B;
    float* sv    = smem + OFF_V;
    float* sbias = smem + OFF_BIAS;
    float* sT    = smem + OFF_W;     // alias: W0 dead after GEMM

    const int t    = threadIdx.x;
    const int lane = t & 31;
    const int wave = t >> 5;

    // ---- Stage shared operands ----
    // Bulk tensors via the Tensor Data Mover (one wave issues 3 DMA descriptors;
    // TENSORcnt tracks completion). data_size=2 (4B).
    if (wave == 0) {
        // W0: 128x128 f32 tile, LDS rows padded 128->144 dwords
        // flags: data_size=2 | pad_enable | pad_interval=6 (128dw) | pad_amount=15 (16dw)
        unsigned flagsW = (2u << 16) | (1u << 20) | (6u << 22) | (15u << 25);
        u32x4 g0w = tdm_group0((unsigned)(unsigned long long)(void*)sW, W0);
        i32x8 g1w = tdm_group1(flagsW, D, D, D, D, D);
        // A, Bm: flat 5120-element 1D copies
        unsigned flagsL = (2u << 16);
        u32x4 g0a = tdm_group0((unsigned)(unsigned long long)(void*)sA, A);
        i32x8 g1a = tdm_group1(flagsL, SZ_A, 1, SZ_A, 0, SZ_A);
        u32x4 g0b = tdm_group0((unsigned)(unsigned long long)(void*)sB, Bm);
        i32x8 g1b = tdm_group1(flagsL, SZ_B, 1, SZ_B, 0, SZ_B);
        i32x4 gz4 = (i32x4)0;
        i32x8 gz8 = (i32x8)0;
        __builtin_amdgcn_tensor_load_to_lds(g0w, g1w, gz4, gz4, gz8, 0);
        __builtin_amdgcn_tensor_load_to_lds(g0a, g1a, gz4, gz4, gz8, 0);
        __builtin_amdgcn_tensor_load_to_lds(g0b, g1b, gz4, gz4, gz8, 0);
        __builtin_amdgcn_s_wait_tensorcnt((short)0);
    }
    if (t < D)    sv[t] = v[t];
    if (t < NHOP) sbias[t] = bvec[t];
    __syncthreads();

    const int lo = lane & 15;     // row-in-tile / col-in-tile
    const int hi = lane >> 4;     // 0/1: K sub-pair select
    const long rowBase = (long)blockIdx.x * ROWS_PER_BLOCK + (long)wave * ROWS_PER_WAVE;
    const long myRow   = rowBase + lo;
    const float* qrow  = q + myRow * D;

    // CDNA5 vmem prefetch (global_prefetch_b8) for final-phase operands
    __builtin_prefetch(r + myRow * D + hi * 64, 0, 0);
    __builtin_prefetch(hop + rowBase, 0, 0);

    // ---- GEMM: T[16x128] = Qtile[16x128] @ W0[128x128] via f32 WMMA ----
    // A-frag (16x4): lanes 0-15 hold M=lo, K={kk,kk+1}; lanes 16-31 K={kk+2,kk+3}
    // B-frag (4x16): VGPR0 = W0[kk+2*hi][col], VGPR1 = W0[kk+2*hi+1][col]
    v8f acc[8];
    #pragma unroll
    for (int tt = 0; tt < 8; ++tt) acc[tt] = (v8f)0.0f;

    #pragma unroll 4
    for (int kk = 0; kk < D; kk += 4) {
        v2f afrag = *(const v2f*)(qrow + kk + 2 * hi);
        const float* wrow = sW + (kk + 2 * hi) * W_STRIDE;
        #pragma unroll
        for (int tt = 0; tt < 8; ++tt) {
            int col = tt * 16 + lo;
            v2f bfrag = { wrow[col], wrow[W_STRIDE + col] };   // ds_load_2addr_b32
            acc[tt] = __builtin_amdgcn_wmma_f32_16x16x4_f32(
                false, afrag, false, bfrag, (short)0, acc[tt], false, false);
        }
    }

    // All waves done reading sW before T tiles overwrite it
    __syncthreads();

    // C/D layout: VGPR i, lanes 0-15 -> (M=i, N=lane); lanes 16-31 -> (M=8+i, N=lane-16)
    float* sTw = sT + wave * (ROWS_PER_WAVE * T_STRIDE);
    #pragma unroll
    for (int tt = 0; tt < 8; ++tt) {
        #pragma unroll
        for (int i = 0; i < 8; ++i) {
            sTw[(i + 8 * hi) * T_STRIDE + tt * 16 + lo] = acc[tt][i];
        }
    }
    __syncthreads();

    // ---- Final phase: 2 lanes per row, 64 columns each ----
    const int   c0 = hi * 64;
    const float* qp = q + myRow * D + c0;
    const float* rp = r + myRow * D + c0;
    const float* tp = sTw + lo * T_STRIDE + c0;
    const int   h  = hop[myRow];
    const float* sAh = sA + h * (D * RHO) + c0 * RHO;
    const float* sBh = sB + h * (D * RHO) + c0 * RHO;

    float basep = 0.0f, hadp = 0.0f;
    float qa[RHO], rb[RHO];
    #pragma unroll
    for (int rr = 0; rr < RHO; ++rr) { qa[rr] = 0.0f; rb[rr] = 0.0f; }

    #pragma unroll 4
    for (int j = 0; j < 64; ++j) {
        float qd = qp[j];
        float rd = rp[j];
        basep = fmaf(tp[j], rd, basep);
        hadp  = fmaf(sv[c0 + j] * qd, rd, hadp);
        #pragma unroll
        for (int rr = 0; rr < RHO; ++rr) {
            qa[rr] = fmaf(qd, sAh[j * RHO + rr], qa[rr]);
            rb[rr] = fmaf(rd, sBh[j * RHO + rr], rb[rr]);
        }
    }

    // combine half-rows across lane^16 (wave32 shuffles)
    basep += __shfl_xor(basep, 16, 32);
    hadp  += __shfl_xor(hadp, 16, 32);
    float delta = 0.0f;
    #pragma unroll
    for (int rr = 0; rr < RHO; ++rr) {
        float qaf = qa[rr] + __shfl_xor(qa[rr], 16, 32);
        float rbf = rb[rr] + __shfl_xor(rb[rr], 16, 32);
        delta = fmaf(qaf, rbf, delta);
    }

    if (hi == 0) {
        float x = basep + delta + hadp + sbias[h];
        out[myRow] = 1.0f / (1.0f + __expf(-x));
    }
}

extern "C" void kernel_launch(void* const* d_in, const int* in_sizes, int n_in,
                              void* d_out, int out_size, void* d_ws, size_t ws_size,
                              hipStream_t stream) {
    const float* q   = (const float*)d_in[0];
    const float* r   = (const float*)d_in[1];
    const int*   hop = (const int*)d_in[2];
    const float* W0  = (const float*)d_in[3];
    const float* A   = (const float*)d_in[4];
    const float* Bm  = (const float*)d_in[5];
    const float* v   = (const float*)d_in[6];
    const float* b   = (const float*)d_in[7];
    float* out = (float*)d_out;

    const int rows = in_sizes[0] / D;                 // 1048576
    const int grid = rows / ROWS_PER_BLOCK;           // 8192
    const size_t smem = (size_t)SMEM_FLOATS * sizeof(float);  // ~115 KB

    DSRQS_fused_kernel<<<grid, 256, smem, stream>>>(q, r, hop, W0, A, Bm, v, b, out);
}